// DMR_FlotRefine_67327907332134
// MI455X (gfx1250) — compile-verified
//
#include <hip/hip_runtime.h>
#include <hip/hip_bf16.h>

typedef __attribute__((ext_vector_type(2))) float v2f;
typedef __attribute__((ext_vector_type(8))) float v8f;

#define BN     2
#define NP     8192
#define KNB    4
#define LROW   (NP * KNB)        // 32768 rows per batch
#define MTOT   (BN * LROW)       // 65536 total rows
#define NGROUP 8
#define EPSGN  1e-5f
#define SLOPE  0.1f

// ---------------------------------------------------------------------------
// sorted-ascending top-4 insertion (strict <, preserves lowest index on ties)
// ---------------------------------------------------------------------------
__device__ __forceinline__ void topk_insert(float v, int j, float (&s)[4], int (&id)[4]) {
  if (v < s[3]) {
    if (v < s[2]) {
      s[3] = s[2]; id[3] = id[2];
      if (v < s[1]) {
        s[2] = s[1]; id[2] = id[1];
        if (v < s[0]) { s[1] = s[0]; id[1] = id[0]; s[0] = v; id[0] = j; }
        else          { s[1] = v;    id[1] = j; }
      } else { s[2] = v; id[2] = j; }
    } else { s[3] = v; id[3] = j; }
  }
}

// ---------------------------------------------------------------------------
// KNN: one wave handles 16 query points; the x·yT part of the distance matrix
// is computed with V_WMMA_F32_16X16X4_F32 (K=4: x,y,z,0). score = |r|^2 - 2*dot
// (the |q|^2 term is row-constant, irrelevant to top-k ordering).
// Two 16-ref tiles per iteration; lanes 0-15 scan tile0, 16-31 scan tile1.
// ---------------------------------------------------------------------------
__global__ __launch_bounds__(32)
void knn_kernel(const float* __restrict__ pc, int* __restrict__ nbr,
                float* __restrict__ edge) {
  __shared__ float tile[2][16][16];
  __shared__ int   midx[16][4];

  const int b    = blockIdx.y;
  const int qt   = blockIdx.x;
  const int lane = threadIdx.x;
  const int half = lane >> 4;
  const int ln   = lane & 15;
  const float* pcb = pc + (size_t)b * NP * 3;

  const int q = qt * 16 + ln;
  const float qx = pcb[q * 3 + 0], qy = pcb[q * 3 + 1], qz = pcb[q * 3 + 2];
  v2f a;
  if (half) { a.x = qz; a.y = 0.0f; } else { a.x = qx; a.y = qy; }

  float bs[4] = {3.0e38f, 3.0e38f, 3.0e38f, 3.0e38f};
  int   bi[4] = {-1, -1, -1, -1};

  for (int t = 0; t < NP / 32; ++t) {
#pragma unroll
    for (int sub = 0; sub < 2; ++sub) {
      const int ridx = t * 32 + sub * 16 + ln;
      const float rx = pcb[ridx * 3 + 0];
      const float ry = pcb[ridx * 3 + 1];
      const float rz = pcb[ridx * 3 + 2];
      const float rn = rx * rx + ry * ry + rz * rz;
      v2f bv;
      if (half) { bv.x = rz; bv.y = 0.0f; } else { bv.x = rx; bv.y = ry; }
      v8f c = {};
      c = __builtin_amdgcn_wmma_f32_16x16x4_f32(false, a, false, bv,
                                                (short)0, c, false, false);
#pragma unroll
      for (int r = 0; r < 8; ++r)
        tile[sub][r + 8 * half][ln] = rn - 2.0f * c[r];
    }
    __syncthreads();
    const int rb = t * 32 + half * 16;
#pragma unroll
    for (int n = 0; n < 16; ++n)
      topk_insert(tile[half][ln][n], rb + n, bs, bi);
    __syncthreads();
  }

  // merge upper-lane lists into lower-lane lists
  if (half) {
#pragma unroll
    for (int j = 0; j < 4; ++j) { tile[0][ln][j] = bs[j]; midx[ln][j] = bi[j]; }
  }
  __syncthreads();
  if (!half) {
#pragma unroll
    for (int j = 0; j < 4; ++j) topk_insert(tile[0][ln][j], midx[ln][j], bs, bi);
    const size_t base = ((size_t)b * NP + q) * KNB;
#pragma unroll
    for (int j = 0; j < 4; ++j) {
      const int nb = bi[j];
      nbr[base + j] = nb;
      edge[(base + j) * 3 + 0] = pcb[nb * 3 + 0] - qx;
      edge[(base + j) * 3 + 1] = pcb[nb * 3 + 1] - qy;
      edge[(base + j) * 3 + 2] = pcb[nb * 3 + 2] - qz;
    }
  }
}

// ---------------------------------------------------------------------------
// Templated GEMM core: 16-row block staged in LDS, fully-unrolled chain of
// V_WMMA_F32_16X16X4_F32 over K, bias add, then wave-level shuffle reduction
// of per-(batch,group) sum/sum^2 before a single atomic per group/tile.
// ---------------------------------------------------------------------------
template <int CinPad, int Cin, int Cout>
__device__ __forceinline__ void mm_core(const float (*Xs)[64],
                                        const float* __restrict__ W,
                                        const float* __restrict__ bias,
                                        int row0, int b, float* __restrict__ Y,
                                        float* __restrict__ statsRaw) {
  const int lane = threadIdx.x;
  const int half = lane >> 4;
  const int ln   = lane & 15;
  constexpr int NT = Cout / 16;
  constexpr int GW = Cout / NGROUP;   // channels (=lanes) per group: 2, 4, or 8
#pragma unroll
  for (int nt = 0; nt < NT; ++nt) {
    const int ch = nt * 16 + ln;
    v8f acc = {};
#pragma unroll
    for (int k0 = 0; k0 < CinPad; k0 += 4) {
      const int kb = k0 + 2 * half;
      v2f av; av.x = Xs[ln][kb]; av.y = Xs[ln][kb + 1];
      v2f bv;
      bv.x = (kb     < Cin) ? W[ch * Cin + kb]     : 0.0f;
      bv.y = (kb + 1 < Cin) ? W[ch * Cin + kb + 1] : 0.0f;
      acc = __builtin_amdgcn_wmma_f32_16x16x4_f32(false, av, false, bv,
                                                  (short)0, acc, false, false);
    }
    const float bb = bias[ch];
    float s = 0.0f, s2 = 0.0f;
#pragma unroll
    for (int r = 0; r < 8; ++r) {
      const float v = acc[r] + bb;
      const int   m = r + 8 * half;
      Y[(size_t)(row0 + m) * Cout + ch] = v;
      s += v; s2 += v * v;
    }
    // reduce across the two half-wave copies of each channel, then across the
    // GW contiguous channels of a group; one atomic writer per group.
    s  += __shfl_xor(s, 16);
    s2 += __shfl_xor(s2, 16);
#pragma unroll
    for (int o = 1; o < GW; o <<= 1) {
      s  += __shfl_xor(s, o);
      s2 += __shfl_xor(s2, o);
    }
    if (half == 0 && (ln % GW) == 0) {
      const int g = ch / GW;
      atomicAdd(&statsRaw[(b * NGROUP + g) * 2 + 0], s);
      atomicAdd(&statsRaw[(b * NGROUP + g) * 2 + 1], s2);
    }
  }
}

// First layer of a SetConv: gather neighbor signal + edge features on the fly.
template <int Cs, int Cout>
__global__ __launch_bounds__(32)
void mm_first_kernel(const float* __restrict__ signal, int sstride,
                     const int* __restrict__ nbr, const float* __restrict__ edge,
                     const float* __restrict__ W, const float* __restrict__ bias,
                     float* __restrict__ Y, float* __restrict__ statsRaw) {
  __shared__ float Xs[16][64];
  constexpr int Cin    = Cs + 3;
  constexpr int CinPad = (Cin + 3) & ~3;
  const int lane = threadIdx.x;
  const int row0 = blockIdx.x * 16;
  const int b    = row0 / LROW;
#pragma unroll
  for (int idx = lane; idx < 16 * CinPad; idx += 32) {
    const int m = idx / CinPad, c = idx % CinPad;
    const int rowg = row0 + m;
    const int l = rowg - b * LROW;
    const int n = l >> 2, j = l & 3;
    float v = 0.0f;
    if (c < Cs) {
      const int nb = nbr[((size_t)b * NP + n) * KNB + j];
      v = signal[((size_t)b * NP + nb) * sstride + c];
    } else if (c < Cin) {
      v = edge[(((size_t)b * NP + n) * KNB + j) * 3 + (c - Cs)];
    }
    Xs[m][c] = v;
  }
  __syncthreads();
  mm_core<CinPad, Cin, Cout>(Xs, W, bias, row0, b, Y, statsRaw);
}

// Middle layer: apply previous layer's GroupNorm + LeakyReLU while staging.
template <int Cin, int Cout>
__global__ __launch_bounds__(32)
void mm_mid_kernel(const float* __restrict__ X,
                   const float* __restrict__ fin, const float* __restrict__ gamma,
                   const float* __restrict__ beta,
                   const float* __restrict__ W, const float* __restrict__ bias,
                   float* __restrict__ Y, float* __restrict__ statsRaw) {
  __shared__ float Xs[16][64];
  const int lane = threadIdx.x;
  const int row0 = blockIdx.x * 16;
  const int b    = row0 / LROW;
#pragma unroll
  for (int idx = lane; idx < 16 * Cin; idx += 32) {
    const int m = idx / Cin, c = idx % Cin;
    const float y = X[(size_t)(row0 + m) * Cin + c];
    const int g = (c * NGROUP) / Cin;
    const float mean = fin[(b * NGROUP + g) * 2 + 0];
    const float rstd = fin[(b * NGROUP + g) * 2 + 1];
    float xn = (y - mean) * rstd * gamma[c] + beta[c];
    Xs[m][c] = (xn >= 0.0f) ? xn : SLOPE * xn;
  }
  __syncthreads();
  mm_core<Cin, Cin, Cout>(Xs, W, bias, row0, b, Y, statsRaw);  // Cin % 4 == 0
}

// ---------------------------------------------------------------------------
// Small support kernels
// ---------------------------------------------------------------------------
__global__ void zero_kernel(float* p, int n) {
  const int i = blockIdx.x * blockDim.x + threadIdx.x;
  if (i < n) p[i] = 0.0f;
}

__global__ void finalize_stats_kernel(const float* __restrict__ raw,
                                      float* __restrict__ fin, float cntInv) {
  const int i = threadIdx.x;
  if (i < BN * NGROUP) {
    const float s = raw[i * 2], s2 = raw[i * 2 + 1];
    const float mean = s * cntInv;
    const float var  = s2 * cntInv - mean * mean;
    fin[i * 2 + 0] = mean;
    fin[i * 2 + 1] = rsqrtf(var + EPSGN);
  }
}

// GN + LeakyReLU + max over k=4 neighbors; pooled rows padded to stride 64.
__global__ void pool_kernel(const float* __restrict__ Yin, int C,
                            const float* __restrict__ fin,
                            const float* __restrict__ gamma,
                            const float* __restrict__ beta,
                            float* __restrict__ pooled) {
  const int idx = blockIdx.x * blockDim.x + threadIdx.x;
  if (idx >= BN * NP * C) return;
  const int c = idx % C;
  const int n = (idx / C) % NP;
  const int b = idx / (C * NP);
  const int g = (c * NGROUP) / C;
  const float mean = fin[(b * NGROUP + g) * 2 + 0];
  const float rstd = fin[(b * NGROUP + g) * 2 + 1];
  const float gm = gamma[c], bt = beta[c];
  float best = -3.0e38f;
#pragma unroll
  for (int j = 0; j < KNB; ++j) {
    const float y = Yin[(((size_t)b * NP + n) * KNB + j) * C + c];
    float xn = (y - mean) * rstd * gm + bt;
    xn = (xn >= 0.0f) ? xn : SLOPE * xn;
    best = fmaxf(best, xn);
  }
  pooled[((size_t)b * NP + n) * 64 + c] = best;
}

__global__ void final_kernel(const float* __restrict__ flow,
                             const float* __restrict__ pooled,
                             const float* __restrict__ Wfc,
                             const float* __restrict__ bfc,
                             float* __restrict__ out) {
  const int idx = blockIdx.x * blockDim.x + threadIdx.x;
  if (idx >= BN * NP) return;
  const float* p = pooled + (size_t)idx * 64;
#pragma unroll
  for (int d = 0; d < 3; ++d) {
    float s = bfc[d];
#pragma unroll
    for (int c = 0; c < 64; ++c) s += Wfc[d * 64 + c] * p[c];
    out[idx * 3 + d] = flow[idx * 3 + d] + s;
  }
}

// ---------------------------------------------------------------------------
// Host launch
// ---------------------------------------------------------------------------
extern "C" void kernel_launch(void* const* d_in, const int* in_sizes, int n_in,
                              void* d_out, int out_size, void* d_ws, size_t ws_size,
                              hipStream_t stream) {
  (void)in_sizes; (void)n_in; (void)out_size; (void)ws_size;
  const float* pc   = (const float*)d_in[0];
  const float* flow = (const float*)d_in[1];
  // params flattened: per layer {W, b, gamma, beta}; conv1 @2, conv2 @14, conv3 @26, fc @38
  const float* P[40];
  for (int i = 2; i < 40; ++i) P[i] = (const float*)d_in[i];

  char* ws = (char*)d_ws;
  int*   nbr    = (int*)  (ws + 0);
  float* edge   = (float*)(ws + 262144);
  float* bufA   = (float*)(ws + 1048576);
  float* bufB   = (float*)(ws + 17825792);
  float* pooled = (float*)(ws + 34603008);
  float* raw    = (float*)(ws + 38797312);   // 9 slots * 32 floats
  float* fin    = (float*)(ws + 38798464);   // 9 slots * 32 floats

  zero_kernel<<<2, 256, 0, stream>>>(raw, 9 * 32);
  knn_kernel<<<dim3(NP / 16, BN), 32, 0, stream>>>(pc, nbr, edge);

  const dim3 mmGrid(MTOT / 16);
  const float inv16 = 1.0f / (LROW * 2.0f);   // C=16: 2 channels/group
  const float inv32 = 1.0f / (LROW * 4.0f);   // C=32
  const float inv64 = 1.0f / (LROW * 8.0f);   // C=64

  // ---- conv1 (3 -> 16) ----
  mm_first_kernel<3, 16><<<mmGrid, 32, 0, stream>>>(flow, 3, nbr, edge, P[2], P[3], bufA, raw + 0);
  finalize_stats_kernel<<<1, 32, 0, stream>>>(raw + 0, fin + 0, inv16);
  mm_mid_kernel<16, 16><<<mmGrid, 32, 0, stream>>>(bufA, fin + 0, P[4], P[5], P[6], P[7], bufB, raw + 32);
  finalize_stats_kernel<<<1, 32, 0, stream>>>(raw + 32, fin + 32, inv16);
  mm_mid_kernel<16, 16><<<mmGrid, 32, 0, stream>>>(bufB, fin + 32, P[8], P[9], P[10], P[11], bufA, raw + 64);
  finalize_stats_kernel<<<1, 32, 0, stream>>>(raw + 64, fin + 64, inv16);
  pool_kernel<<<(BN * NP * 16 + 255) / 256, 256, 0, stream>>>(bufA, 16, fin + 64, P[12], P[13], pooled);

  // ---- conv2 (16 -> 32) ----
  mm_first_kernel<16, 32><<<mmGrid, 32, 0, stream>>>(pooled, 64, nbr, edge, P[14], P[15], bufB, raw + 96);
  finalize_stats_kernel<<<1, 32, 0, stream>>>(raw + 96, fin + 96, inv32);
  mm_mid_kernel<32, 32><<<mmGrid, 32, 0, stream>>>(bufB, fin + 96, P[16], P[17], P[18], P[19], bufA, raw + 128);
  finalize_stats_kernel<<<1, 32, 0, stream>>>(raw + 128, fin + 128, inv32);
  mm_mid_kernel<32, 32><<<mmGrid, 32, 0, stream>>>(bufA, fin + 128, P[20], P[21], P[22], P[23], bufB, raw + 160);
  finalize_stats_kernel<<<1, 32, 0, stream>>>(raw + 160, fin + 160, inv32);
  pool_kernel<<<(BN * NP * 32 + 255) / 256, 256, 0, stream>>>(bufB, 32, fin + 160, P[24], P[25], pooled);

  // ---- conv3 (32 -> 64) ----
  mm_first_kernel<32, 64><<<mmGrid, 32, 0, stream>>>(pooled, 64, nbr, edge, P[26], P[27], bufA, raw + 192);
  finalize_stats_kernel<<<1, 32, 0, stream>>>(raw + 192, fin + 192, inv64);
  mm_mid_kernel<64, 64><<<mmGrid, 32, 0, stream>>>(bufA, fin + 192, P[28], P[29], P[30], P[31], bufB, raw + 224);
  finalize_stats_kernel<<<1, 32, 0, stream>>>(raw + 224, fin + 224, inv64);
  mm_mid_kernel<64, 64><<<mmGrid, 32, 0, stream>>>(bufB, fin + 224, P[32], P[33], P[34], P[35], bufA, raw + 256);
  finalize_stats_kernel<<<1, 32, 0, stream>>>(raw + 256, fin + 256, inv64);
  pool_kernel<<<(BN * NP * 64 + 255) / 256, 256, 0, stream>>>(bufA, 64, fin + 256, P[36], P[37], pooled);

  // ---- fc + residual ----
  final_kernel<<<(BN * NP + 255) / 256, 256, 0, stream>>>(flow, pooled, P[38], P[39], (float*)d_out);
}